// HierarchicalAttentionPooling_12670153523236
// MI455X (gfx1250) — compile-verified
//
#include <hip/hip_runtime.h>
#include <hip/hip_bf16.h>

// Problem constants (match reference)
#define B_  8
#define N_  8192
#define D_  512
#define H_  8
#define S_  16
#define DH_ 64
#define SCALE_ 0.125f   // 1/sqrt(64)

typedef float v2f __attribute__((ext_vector_type(2)));
typedef float v8f __attribute__((ext_vector_type(8)));

// Workspace layout (float offsets). Total ~4.93 MB.
#define OFF_QH     0         // 512
#define OFF_WK     512       // H*D = 4096 : wk_eff[h][d]
#define OFF_CK     4608      // 8
#define OFF_WKB    4616      // 128 chunks * 32 lanes * 2 = 8192 (WMMA B fragments, 16B aligned)
#define OFF_SCORES 12808     // B*H*N = 524288
#define OFF_M      537096    // B*S*H = 1024
#define OFF_L      538120    // 1024
#define OFF_XBAR   539144    // B*S*H*D = 524288
#define OFF_SCTX   1063432   // B*S*D = 65536
#define OFF_SPOOL  1128968   // B*S*D = 65536
#define OFF_SPBAR  1194504   // B*H*D = 32768
#define OFF_CTX2   1227272   // B*D = 4096

__device__ __forceinline__ float wave_max(float v) {
  #pragma unroll
  for (int m = 16; m >= 1; m >>= 1) v = fmaxf(v, __shfl_xor(v, m, 32));
  return v;
}
__device__ __forceinline__ float wave_sum(float v) {
  #pragma unroll
  for (int m = 16; m >= 1; m >>= 1) v += __shfl_xor(v, m, 32);
  return v;
}

// ---------------------------------------------------------------------------
// K0: qh = query@Wq.T + bq; wk_eff[h] = Wk_h.T @ qh[h]; ck[h] = qh[h]·bk_h;
//     wkB = wk_eff swizzled into per-lane WMMA-B fragments (16 cols, 8 pad=0).
// ---------------------------------------------------------------------------
__global__ void k0_prep(const float* __restrict__ query,
                        const float* __restrict__ w,     // in_proj_w (3D x D)
                        const float* __restrict__ bias,  // in_proj_b (3D)
                        float* __restrict__ ws) {
  float* qh  = ws + OFF_QH;
  float* wk  = ws + OFF_WK;
  float* ck  = ws + OFF_CK;
  float* wkB = ws + OFF_WKB;
  const int tid = threadIdx.x;  // 256 threads

  for (int d = tid; d < D_; d += 256) {
    float acc = bias[d];
    #pragma unroll 4
    for (int j = 0; j < D_; ++j) acc += query[j] * w[(size_t)d * D_ + j];
    qh[d] = acc;
  }
  __syncthreads();

  for (int idx = tid; idx < H_ * D_; idx += 256) {
    int h = idx / D_, d = idx % D_;
    float acc = 0.f;
    #pragma unroll 4
    for (int j = 0; j < DH_; ++j)
      acc += qh[h * DH_ + j] * w[((size_t)(D_ + h * DH_ + j)) * D_ + d];
    wk[idx] = acc;
  }
  if (tid < H_) {
    float acc = 0.f;
    for (int j = 0; j < DH_; ++j) acc += qh[tid * DH_ + j] * bias[D_ + tid * DH_ + j];
    ck[tid] = acc;
  }
  __syncthreads();

  // B fragment layout (4x16 f32): VGPR0 = row K=(lane>>4), VGPR1 = row K=2+(lane>>4)
  for (int idx = tid; idx < (D_ / 4) * 32; idx += 256) {
    int kk = idx >> 5, L = idx & 31;
    int n = L & 15, kd = 4 * kk + (L >> 4);
    wkB[2 * idx]     = (n < H_) ? wk[n * D_ + kd]     : 0.f;
    wkB[2 * idx + 1] = (n < H_) ? wk[n * D_ + kd + 2] : 0.f;
  }
}

// ---------------------------------------------------------------------------
// K1: scores[b,h,n] = SCALE*(wk_eff[h]·x[b,n] + ck[h]); mask -> -inf.
// wkB staged once per block in LDS (ds_load_b64 in the hot loop); each wave
// computes TWO 16-row M-tiles sharing each B fragment:
//   per chunk: 2x global_load_b64 (x) + 1x ds_load_b64 (wkB) + 2x WMMA.
// ---------------------------------------------------------------------------
__global__ void k1_scores(const float* __restrict__ x,
                          const float* __restrict__ ws,
                          const unsigned char* __restrict__ mask,
                          float* __restrict__ scores) {
  __shared__ float lwkB[(D_ / 4) * 64];  // 8192 floats = 32 KB

  // Cooperative stage: global -> LDS (b128 copies)
  {
    const float4* src = (const float4*)(ws + OFF_WKB);
    float4* dst = (float4*)lwkB;
    for (int i = threadIdx.x; i < (D_ / 4) * 16; i += 128) dst[i] = src[i];
  }
  __syncthreads();

  const float* ck = ws + OFF_CK;
  const int lane = threadIdx.x & 31;
  const int tile = blockIdx.x * 4 + (threadIdx.x >> 5);  // tile = 32 rows
  const int b  = tile / (N_ / 32);
  const int n0 = (tile % (N_ / 32)) * 32;

  // A fragment (16x4 f32): lane holds row M=(lane&15), K pair base 2*(lane>>4)
  const float* xrow0 = x + ((size_t)b * N_ + n0 + (lane & 15)) * D_ + 2 * (lane >> 4);
  const float* xrow1 = xrow0 + (size_t)16 * D_;
  const float* bptr  = lwkB + lane * 2;

  v8f c0 = {0.f, 0.f, 0.f, 0.f, 0.f, 0.f, 0.f, 0.f};
  v8f c1 = {0.f, 0.f, 0.f, 0.f, 0.f, 0.f, 0.f, 0.f};
  #pragma unroll 2
  for (int kk = 0; kk < D_ / 4; ++kk) {
    v2f bf = *(const v2f*)(bptr + kk * 64);     // LDS
    v2f a0 = *(const v2f*)(xrow0 + 4 * kk);     // global
    v2f a1 = *(const v2f*)(xrow1 + 4 * kk);     // global
    c0 = __builtin_amdgcn_wmma_f32_16x16x4_f32(false, a0, false, bf, (short)0, c0,
                                               false, false);
    c1 = __builtin_amdgcn_wmma_f32_16x16x4_f32(false, a1, false, bf, (short)0, c1,
                                               false, false);
  }

  const int h = lane & 15;       // column = head
  if (h < H_) {
    const float ckh = ck[h];
    const int mo = (lane >> 4) * 8;  // C rows: r + 8*(lane>=16)
    #pragma unroll
    for (int r = 0; r < 8; ++r) {
      int pos0 = n0 + r + mo;
      float v0 = (c0[r] + ckh) * SCALE_;
      if (!mask[(size_t)b * N_ + pos0]) v0 = -__builtin_inff();
      scores[((size_t)b * H_ + h) * N_ + pos0] = v0;
      int pos1 = pos0 + 16;
      float v1 = (c1[r] + ckh) * SCALE_;
      if (!mask[(size_t)b * N_ + pos1]) v1 = -__builtin_inff();
      scores[((size_t)b * H_ + h) * N_ + pos1] = v1;
    }
  }
}

// ---------------------------------------------------------------------------
// K2: per (b,s,h) softmax max & sum over segment [start,end)
// ---------------------------------------------------------------------------
__global__ void k2_stats(const float* __restrict__ scores,
                         const int* __restrict__ sb,
                         float* __restrict__ mArr, float* __restrict__ lArr) {
  const int lane = threadIdx.x & 31;
  const int task = blockIdx.x * 8 + (threadIdx.x >> 5);  // ((b*S+s)*H + h)
  const int b = task / (S_ * H_), s = (task / H_) % S_, h = task % H_;
  const int start = sb[(b * S_ + s) * 2], end = sb[(b * S_ + s) * 2 + 1];
  const float* sc = scores + ((size_t)b * H_ + h) * N_;

  float m = -__builtin_inff();
  for (int n = start + lane; n < end; n += 32) m = fmaxf(m, sc[n]);
  m = wave_max(m);
  float l = 0.f;
  for (int n = start + lane; n < end; n += 32) l += __expf(sc[n] - m);
  l = wave_sum(l);
  if (lane == 0) { mArr[task] = m; lArr[task] = l; }
}

// ---------------------------------------------------------------------------
// K3: xbar[b,s,h,d] = sum_n softmax(scores)·x[b,n,d] via WMMA.
// A = probs (rows = 16 heads, 8 pad rows zeroed via invl=0), B = x columns.
// Branchless clamp+select keeps EXEC all-1s for WMMA; handles any seg length.
// ---------------------------------------------------------------------------
__global__ void k3_xbar(const float* __restrict__ x,
                        const float* __restrict__ scores,
                        const int* __restrict__ sb,
                        const float* __restrict__ mArr,
                        const float* __restrict__ lArr,
                        float* __restrict__ xbar) {
  const int lane = threadIdx.x & 31;
  const int tile = blockIdx.x * 4 + (threadIdx.x >> 5);  // B*S*32 tiles
  const int d0 = (tile & 31) * 16;
  const int bs = tile >> 5;
  const int b = bs / S_;
  const int start = sb[bs * 2], end = sb[bs * 2 + 1];

  const int hrow = lane & 15;                 // A row = head
  const int hq = hrow < H_ ? hrow : H_ - 1;   // clamp for safe loads
  const float mh   = mArr[bs * H_ + hq];
  const float invl = (hrow < H_) ? 1.0f / lArr[bs * H_ + hq] : 0.f;
  const float* srow = scores + ((size_t)b * H_ + hq) * N_;
  const float* xb = x + (size_t)b * N_ * D_ + d0 + (lane & 15);

  const int klim = end - start;
  const int chunks = (klim + 3) >> 2;
  const int ka = 2 * (lane >> 4);
  const int kb = (lane >> 4);

  v8f c = {0.f, 0.f, 0.f, 0.f, 0.f, 0.f, 0.f, 0.f};
  for (int ci = 0; ci < chunks; ++ci) {
    // A fragment: probs p[hrow][k], k = 4*ci + ka + {0,1}
    int n0p = start + 4 * ci + ka, n1p = n0p + 1;
    int n0c = n0p < end ? n0p : end - 1;
    int n1c = n1p < end ? n1p : end - 1;
    float p0 = __expf(srow[n0c] - mh) * invl;
    float p1 = __expf(srow[n1c] - mh) * invl;
    v2f a; a.x = (n0p < end) ? p0 : 0.f; a.y = (n1p < end) ? p1 : 0.f;
    // B fragment: x rows k = 4*ci + kb and +2, column d0+(lane&15)
    int m0 = start + 4 * ci + kb, m1 = m0 + 2;
    int m0c = m0 < end ? m0 : end - 1;
    int m1c = m1 < end ? m1 : end - 1;
    float b0 = xb[(size_t)m0c * D_];
    float b1 = xb[(size_t)m1c * D_];
    v2f bf; bf.x = (m0 < end) ? b0 : 0.f; bf.y = (m1 < end) ? b1 : 0.f;
    c = __builtin_amdgcn_wmma_f32_16x16x4_f32(false, a, false, bf, (short)0, c,
                                              false, false);
  }

  // C: lanes 0-15 hold head rows 0..7 (rows 8..15 are zero pad on lanes 16-31)
  if (lane < 16) {
    #pragma unroll
    for (int r = 0; r < 8; ++r)
      xbar[((size_t)bs * H_ + r) * D_ + d0 + lane] = c[r];
  }
}

// ---------------------------------------------------------------------------
// K4a: step_ctx[b,s,f] = Wv_row(f)·xbar[b,s,h(f)] + bv[f]   (f = h*64+j)
// ---------------------------------------------------------------------------
__global__ void k4a_stepctx(const float* __restrict__ w,
                            const float* __restrict__ bias,
                            const float* __restrict__ xbar,
                            float* __restrict__ sctx) {
  const int idx = blockIdx.x * blockDim.x + threadIdx.x;  // B*S*D
  const int f = idx % D_, bs = idx / D_, h = f / DH_;
  const float* wr = w + ((size_t)(2 * D_ + f)) * D_;
  const float* xv = xbar + ((size_t)bs * H_ + h) * D_;
  float acc = bias[2 * D_ + f];
  #pragma unroll 4
  for (int d = 0; d < D_; ++d) acc += wr[d] * xv[d];
  sctx[idx] = acc;
}

// K4b: step_pooled[b,s,e] = out_w[e]·step_ctx[b,s] + out_b[e]
__global__ void k4b_pool(const float* __restrict__ outw,
                         const float* __restrict__ outb,
                         const float* __restrict__ sctx,
                         float* __restrict__ spool) {
  const int idx = blockIdx.x * blockDim.x + threadIdx.x;  // B*S*D
  const int e = idx % D_, bs = idx / D_;
  const float* wr = outw + (size_t)e * D_;
  const float* cv = sctx + (size_t)bs * D_;
  float acc = outb[e];
  #pragma unroll 4
  for (int d = 0; d < D_; ++d) acc += wr[d] * cv[d];
  spool[idx] = acc;
}

// ---------------------------------------------------------------------------
// K5a: level-2 attention over S=16 pooled steps; reuses wk_eff / ck.
//      spbar[b,h,d] = sum_s softmax_s(SCALE*(wk[h]·sp[b,s]+ck[h])) * sp[b,s,d]
// ---------------------------------------------------------------------------
__global__ void k5a_attn2(const float* __restrict__ wk,
                          const float* __restrict__ ck,
                          const float* __restrict__ spool,
                          float* __restrict__ spbar) {
  const int lane = threadIdx.x & 31;
  const int task = blockIdx.x * 8 + (threadIdx.x >> 5);  // b*H + h
  const int b = task / H_, h = task % H_;

  float scv[S_];
  for (int s = 0; s < S_; ++s) {
    const float* sp = spool + ((size_t)b * S_ + s) * D_;
    float part = 0.f;
    for (int d = lane; d < D_; d += 32) part += wk[h * D_ + d] * sp[d];
    part = wave_sum(part);
    scv[s] = (part + ck[h]) * SCALE_;
  }
  float mx = -__builtin_inff();
  #pragma unroll
  for (int s = 0; s < S_; ++s) mx = fmaxf(mx, scv[s]);
  float a[S_]; float sum = 0.f;
  #pragma unroll
  for (int s = 0; s < S_; ++s) { a[s] = __expf(scv[s] - mx); sum += a[s]; }
  const float inv = 1.0f / sum;

  for (int d = lane; d < D_; d += 32) {
    float acc = 0.f;
    #pragma unroll
    for (int s = 0; s < S_; ++s)
      acc += a[s] * inv * spool[((size_t)b * S_ + s) * D_ + d];
    spbar[(size_t)task * D_ + d] = acc;
  }
}

// K5b: ctx2[b,f] = Wv_row(f)·spbar[b,h(f)] + bv[f]
__global__ void k5b_ctx2(const float* __restrict__ w,
                         const float* __restrict__ bias,
                         const float* __restrict__ spbar,
                         float* __restrict__ ctx2) {
  const int idx = blockIdx.x * blockDim.x + threadIdx.x;  // B*D
  const int b = idx / D_, f = idx % D_, h = f / DH_;
  const float* wr = w + ((size_t)(2 * D_ + f)) * D_;
  const float* sv = spbar + ((size_t)(b * H_ + h)) * D_;
  float acc = bias[2 * D_ + f];
  #pragma unroll 4
  for (int d = 0; d < D_; ++d) acc += wr[d] * sv[d];
  ctx2[idx] = acc;
}

// K5c: out[b,e] = out_w[e]·ctx2[b] + out_b[e]
__global__ void k5c_out(const float* __restrict__ outw,
                        const float* __restrict__ outb,
                        const float* __restrict__ ctx2,
                        float* __restrict__ out) {
  const int idx = blockIdx.x * blockDim.x + threadIdx.x;  // B*D
  const int b = idx / D_, e = idx % D_;
  const float* wr = outw + (size_t)e * D_;
  const float* cv = ctx2 + (size_t)b * D_;
  float acc = outb[e];
  #pragma unroll 4
  for (int d = 0; d < D_; ++d) acc += wr[d] * cv[d];
  out[idx] = acc;
}

extern "C" void kernel_launch(void* const* d_in, const int* in_sizes, int n_in,
                              void* d_out, int out_size, void* d_ws, size_t ws_size,
                              hipStream_t stream) {
  const float* x     = (const float*)d_in[0];
  const float* query = (const float*)d_in[1];
  const float* ipw   = (const float*)d_in[2];
  const float* ipb   = (const float*)d_in[3];
  const float* outw  = (const float*)d_in[4];
  const float* outb  = (const float*)d_in[5];
  const unsigned char* mask = (const unsigned char*)d_in[6];
  const int* sb      = (const int*)d_in[7];
  float* out = (float*)d_out;
  float* ws  = (float*)d_ws;
  (void)in_sizes; (void)n_in; (void)out_size; (void)ws_size;

  k0_prep<<<1, 256, 0, stream>>>(query, ipw, ipb, ws);
  // 2048 tiles of 32 rows, 4 waves/block -> 512 blocks
  k1_scores<<<(B_ * (N_ / 32)) / 4, 128, 0, stream>>>(x, ws, mask, ws + OFF_SCORES);
  k2_stats<<<(B_ * S_ * H_) / 8, 256, 0, stream>>>(ws + OFF_SCORES, sb,
                                                   ws + OFF_M, ws + OFF_L);
  k3_xbar<<<(B_ * S_ * 32) / 4, 128, 0, stream>>>(x, ws + OFF_SCORES, sb,
                                                  ws + OFF_M, ws + OFF_L,
                                                  ws + OFF_XBAR);
  k4a_stepctx<<<(B_ * S_ * D_) / 256, 256, 0, stream>>>(ipw, ipb, ws + OFF_XBAR,
                                                        ws + OFF_SCTX);
  k4b_pool<<<(B_ * S_ * D_) / 256, 256, 0, stream>>>(outw, outb, ws + OFF_SCTX,
                                                     ws + OFF_SPOOL);
  k5a_attn2<<<(B_ * H_) / 8, 256, 0, stream>>>(ws + OFF_WK, ws + OFF_CK,
                                               ws + OFF_SPOOL, ws + OFF_SPBAR);
  k5b_ctx2<<<(B_ * D_) / 256, 256, 0, stream>>>(ipw, ipb, ws + OFF_SPBAR,
                                                ws + OFF_CTX2);
  k5c_out<<<(B_ * D_) / 256, 256, 0, stream>>>(outw, outb, ws + OFF_CTX2, out);
}